// PointPillarsScatter_trt_88845693485727
// MI455X (gfx1250) — compile-verified
//
#include <hip/hip_runtime.h>
#include <hip/hip_bf16.h>
#include <stdint.h>

#define C_FEAT 64
#define P_PILL 100000
#define NX_DIM 512
#define NXY (512 * 512)
#define TP 32            // pillars per block (100000 / 32 = 3125 exactly)

// ---------------------------------------------------------------------------
// Kernel 1: zero the 64 MB canvas with 128-bit stores (global_store_b128).
// 16,777,216 floats = 4,194,304 float4 -> 16384 blocks x 256 threads.
// ---------------------------------------------------------------------------
__global__ void pps_zero_canvas(float4* __restrict__ out) {
    const size_t i = (size_t)blockIdx.x * blockDim.x + threadIdx.x;
    out[i] = make_float4(0.f, 0.f, 0.f, 0.f);
}

// ---------------------------------------------------------------------------
// Kernel 2: masked scatter of feature columns.
// Block handles TP=32 pillars with 256 threads (8 waves).
//   - Stage the block's 96 coord dwords into LDS via the CDNA5 async
//     global->LDS engine (ASYNCcnt), then compute masked flat indices once.
//   - Wave w covers channels c = 8w..8w+7; lane = pillar within block.
//     feat[c*P + p] reads are 128B-coalesced per wave; canvas writes are
//     4B scatters into a 64MB region that is L2-resident on MI455X.
// ---------------------------------------------------------------------------
__global__ void pps_scatter(const float* __restrict__ feat,
                            const int*   __restrict__ coords,
                            const int*   __restrict__ mask,
                            float* __restrict__ out) {
    __shared__ int s_coords[TP * 3];
    __shared__ int s_idx[TP];      // -1 => masked off

    const int tid = threadIdx.x;
    const int p0  = blockIdx.x * TP;

    // --- async stage coords[p0*3 .. p0*3+95] into LDS (one b32 per lane) ---
    if (tid < TP * 3) {
        unsigned lds_off = (unsigned)(uintptr_t)(&s_coords[tid]);   // low 32b of
                                                                    // generic ptr
                                                                    // = LDS offset
        unsigned long long gaddr =
            (unsigned long long)(uintptr_t)(coords + (size_t)p0 * 3 + tid);
        asm volatile("global_load_async_to_lds_b32 %0, %1, off"
                     :: "v"(lds_off), "v"(gaddr)
                     : "memory");
    }
    asm volatile("s_wait_asynccnt 0x0" ::: "memory");
    __syncthreads();

    // --- compute masked flat canvas index per pillar (first wave only) ---
    if (tid < TP) {
        const int p = p0 + tid;
        int idx = -1;
        if (mask[p] != 0) {
            const int y = s_coords[tid * 3 + 1];
            const int x = s_coords[tid * 3 + 2];
            idx = y * NX_DIM + x;
        }
        s_idx[tid] = idx;
    }
    __syncthreads();

    // --- scatter: wave 'cgrp' handles 8 channels for 32 pillars ---
    const int lane = tid & 31;      // pillar within block
    const int cgrp = tid >> 5;      // 0..7
    const int p    = p0 + lane;
    const int idx  = s_idx[lane];
    if (idx >= 0) {
        #pragma unroll
        for (int i = 0; i < 8; ++i) {
            const int c = cgrp * 8 + i;
            out[(size_t)c * NXY + idx] = feat[(size_t)c * P_PILL + p];
        }
    }
}

// ---------------------------------------------------------------------------
// Host-side launcher (graph-capture safe: only kernel launches on `stream`).
// Inputs (setup_inputs order):
//   d_in[0] = voxel_features  float32 [64, 100000]
//   d_in[1] = coords          int     [100000, 3]   (b, y, x)
//   d_in[2] = voxel_mask      int     [1, 100000]
// Output: d_out = canvas float32 [1, 64, 512, 512]
// ---------------------------------------------------------------------------
extern "C" void kernel_launch(void* const* d_in, const int* in_sizes, int n_in,
                              void* d_out, int out_size, void* d_ws, size_t ws_size,
                              hipStream_t stream) {
    const float* feat   = (const float*)d_in[0];
    const int*   coords = (const int*)d_in[1];
    const int*   mask   = (const int*)d_in[2];
    float*       out    = (float*)d_out;

    // Zero-fill: 64*512*512 floats / 4 per thread / 256 per block = 16384 blocks
    {
        const int total_f4 = (C_FEAT * NXY) / 4;
        const int threads  = 256;
        const int blocks   = total_f4 / threads;   // 16384
        pps_zero_canvas<<<blocks, threads, 0, stream>>>((float4*)out);
    }

    // Scatter: 100000 pillars / 32 per block = 3125 blocks
    {
        const int threads = 256;
        const int blocks  = P_PILL / TP;           // 3125
        pps_scatter<<<blocks, threads, 0, stream>>>(feat, coords, mask, out);
    }
}